// LPEBuffer_82712480186778
// MI455X (gfx1250) — compile-verified
//
#include <hip/hip_runtime.h>
#include <math.h>

// Sizes from the reference
#define BSZ   64
#define CAPQ  131072
#define DIM   768
#define NTOT  (CAPQ + 3*BSZ)   // 131264 columns of pre_embeds
#define NROW  (2*BSZ)          // 128 query rows
#define THRF  0.5f
#define LDST  772              // 768 + 4 pad floats -> LDS bank spread

typedef __attribute__((ext_vector_type(2))) float v2f;
typedef __attribute__((ext_vector_type(8))) float v8f;

// order-preserving float <-> uint encoding for atomicMin/Max
__device__ __forceinline__ unsigned fenc(float f) {
  unsigned u = __float_as_uint(f);
  return (u & 0x80000000u) ? ~u : (u | 0x80000000u);
}
__device__ __forceinline__ float fdec(unsigned u) {
  unsigned b = (u & 0x80000000u) ? (u ^ 0x80000000u) : ~u;
  return __uint_as_float(b);
}

// ---------------------------------------------------------------------------
// Kernel 1: zero/seed accumulators, q-row norms, W column-difference
// ---------------------------------------------------------------------------
__global__ void init_kernel(const float* __restrict__ vl,
                            const float* __restrict__ W,
                            const float* __restrict__ b,
                            double* S1, double* SL, double* S0, double* Lsum,
                            float* qnorm, float* wd, float* bdiff,
                            unsigned* minU, unsigned* maxU, unsigned* cnt) {
  int t = threadIdx.x;
  if (t < NROW) {
    S1[t] = 0.0; SL[t] = 0.0; S0[t] = 0.0;
    minU[t] = 0xFFFFFFFFu; maxU[t] = 0u; cnt[t] = 0u;
    const float* q = vl + (size_t)(BSZ + t) * DIM;
    float s = 0.f;
    for (int k = 0; k < DIM; ++k) s = fmaf(q[k], q[k], s);
    qnorm[t] = sqrtf(s);
  }
  for (int k = t; k < DIM; k += blockDim.x) wd[k] = W[2*k + 1] - W[2*k];
  if (t == 0) { *Lsum = 0.0; *bdiff = b[1] - b[0]; }
}

// ---------------------------------------------------------------------------
// Kernel 2: fused  (queue scores + norms + 128xN cosine GEMM + row reductions)
// One block = one 16-column stripe; 8 waves each own a 16x16 WMMA tile.
// ---------------------------------------------------------------------------
__global__ __launch_bounds__(256)
void sim_kernel(const float* __restrict__ vl,  const float* __restrict__ score,
                const float* __restrict__ itm, const float* __restrict__ exq,
                const float* __restrict__ exl,
                const float* __restrict__ qnorm, const float* __restrict__ wd,
                const float* __restrict__ bdiff,
                double* __restrict__ S1a, double* __restrict__ SLa,
                double* __restrict__ S0a, double* __restrict__ Lsum,
                unsigned* __restrict__ minU, unsigned* __restrict__ maxU,
                unsigned* __restrict__ cnt) {
  __shared__ float pe[16 * LDST];                 // 16 columns of pre_embeds
  __shared__ float pnorm[16], pscore[16], plabel[16];
  __shared__ float qn[NROW], rs[NROW];

  const int tid = threadIdx.x;
  const int j0  = blockIdx.x * 16;

  // ---- stage 16 pre_embeds rows into LDS (coalesced float4) ----
  for (int idx = tid; idx < 16 * (DIM/4); idx += 256) {
    int c  = idx / (DIM/4);
    int kq = idx - c * (DIM/4);
    int j  = j0 + c;
    const float* src = (j < 3*BSZ) ? (vl  + (size_t)j * DIM)
                                   : (exq + (size_t)(j - 3*BSZ) * DIM);
    *(float4*)&pe[c * LDST + 4*kq] = *(const float4*)(src + 4*kq);
  }
  if (tid < NROW) { qn[tid] = qnorm[tid]; rs[tid] = score[BSZ + tid]; }
  __syncthreads();

  // ---- per-column: norm + queue score (softmax2 == sigmoid(l1-l0)) ----
  {
    int c = tid >> 4, t = tid & 15;
    const float* col = &pe[c * LDST];
    float n2 = 0.f, ds = 0.f;
    for (int k = t; k < DIM; k += 16) {
      float v = col[k];
      n2 = fmaf(v, v, n2);
      ds = fmaf(v, wd[k], ds);
    }
    for (int m = 8; m >= 1; m >>= 1) {
      n2 += __shfl_xor(n2, m, 32);
      ds += __shfl_xor(ds, m, 32);
    }
    if (t == 0) {
      int j = j0 + c;
      pnorm[c]  = sqrtf(n2);
      pscore[c] = (j < 3*BSZ) ? score[j]
                              : (1.0f / (1.0f + expf(-(ds + *bdiff))));
    }
  }
  if (tid < 16) {
    int j = j0 + tid;
    plabel[tid] = (j < 3*BSZ) ? itm[j] : exl[j - 3*BSZ];
  }
  __syncthreads();

  // ---- global label sum (each column counted exactly once, by its block) ----
  if (tid < 16) {
    double lv = (double)plabel[tid];
    for (int m = 8; m >= 1; m >>= 1) lv += __shfl_xor(lv, m, 32);
    if (tid == 0) atomicAdd(Lsum, lv);
  }

  // ---- 16x16xK fp32 WMMA: D = q_tile * pe_tile^T ----
  const int l    = tid & 31;      // lane
  const int w    = tid >> 5;      // wave -> rows 16w..16w+15
  const int col  = l & 15;        // B column / A row-within-tile
  const int half = l >> 4;
  const int koff = half * 2;      // ISA 16x4 A / 4x16 B layout: contiguous K pair

  const float* ap = vl + (size_t)(BSZ + 16*w + col) * DIM + koff;
  const float* bp = &pe[col * LDST + koff];

  v8f acc = {};
#pragma unroll 8
  for (int k0 = 0; k0 < DIM; k0 += 4) {
    v2f a  = *(const v2f*)(ap + k0);
    v2f bb = *(const v2f*)(bp + k0);
    acc = __builtin_amdgcn_wmma_f32_16x16x4_f32(
        false, a, false, bb, (short)0, acc, false, false);
  }

  // ---- epilogue: sim / filter / per-row reductions over 16 columns ----
  const float pn = pnorm[col];
  const float ps = pscore[col];
  const float lb = plabel[col];
#pragma unroll
  for (int v = 0; v < 8; ++v) {
    int row = 16*w + v + (half << 3);
    float dot = acc[v];
    float cosv = dot / fmaxf(qn[row] * pn, 1e-8f);
    float dd = ps - rs[row];
    float sim = cosv * (1.0f - dd * dd);
    bool  filt = (cosv < THRF);
    float s = filt ? -1.0f : sim;

    float  mn = s, mx = s;
    double s1 = (double)s;
    double sl = (double)s * (double)lb;
    double s0 = filt ? 0.0 : (double)sim;
    int    cc = filt ? 0 : 1;
    for (int m = 8; m >= 1; m >>= 1) {
      mn  = fminf(mn, __shfl_xor(mn, m, 32));
      mx  = fmaxf(mx, __shfl_xor(mx, m, 32));
      s1 += __shfl_xor(s1, m, 32);
      sl += __shfl_xor(sl, m, 32);
      s0 += __shfl_xor(s0, m, 32);
      cc += __shfl_xor(cc, m, 32);
    }
    if (col == 0) {
      atomicMin(&minU[row], fenc(mn));
      atomicMax(&maxU[row], fenc(mx));
      atomicAdd(&S1a[row], s1);
      atomicAdd(&SLa[row], sl);
      atomicAdd(&S0a[row], s0);
      atomicAdd(&cnt[row], (unsigned)cc);
    }
  }
}

// ---------------------------------------------------------------------------
// Kernel 3: finalize the 577 outputs
// ---------------------------------------------------------------------------
__global__ void fin_kernel(const double* __restrict__ S1a,
                           const double* __restrict__ SLa,
                           const double* __restrict__ S0a,
                           const double* __restrict__ Lsum,
                           const unsigned* __restrict__ minU,
                           const unsigned* __restrict__ maxU,
                           const unsigned* __restrict__ cnt,
                           float* __restrict__ out) {
  int t = threadIdx.x;
  if (t >= 3*BSZ) return;
  float* o_itm = out;                  // exp_itm_label       [192]
  float* o_wo  = out + 3*BSZ;          // exp_wo_alter_label  [192]
  float* o_g   = out + 6*BSZ;          // gamma               [1]
  float* o_w   = out + 6*BSZ + 1;      // weights             [192]
  if (t == 0) *o_g = (float)((double)NTOT / (double)(CAPQ + 3*BSZ));
  if (t < BSZ) { o_itm[t] = 1.f; o_wo[t] = 1.f; o_w[t] = 0.f; return; }

  int r = t - BSZ;
  double smin = (double)fdec(minU[r]);
  double smax = (double)fdec(maxU[r]);
  double den  = smax - smin + 1e-8;
  double sumd  = (S1a[r] - (double)NTOT * smin) / den;   // sum of distr numer
  double sumdl = (SLa[r] - smin * (*Lsum)) / den;        // sum of distr*label
  double wo  = sumdl / (sumd + 1e-8);
  double alt = wo > 0.0 ? wo : 0.0;
  unsigned c = cnt[r];
  double wv = c ? (S0a[r] / (double)c) : 0.0;
  double wt = wv - 0.5; if (wt < 0.0) wt = 0.0; wt *= 2.0;   // /(1-THR)
  o_itm[t] = (float)alt;
  o_wo[t]  = (float)wo;
  o_w[t]   = (float)wt;
}

// ---------------------------------------------------------------------------
extern "C" void kernel_launch(void* const* d_in, const int* in_sizes, int n_in,
                              void* d_out, int out_size, void* d_ws, size_t ws_size,
                              hipStream_t stream) {
  (void)in_sizes; (void)n_in; (void)out_size; (void)ws_size;
  const float* vl  = (const float*)d_in[0];
  const float* sc  = (const float*)d_in[1];
  const float* itm = (const float*)d_in[2];
  const float* exq = (const float*)d_in[3];
  const float* exl = (const float*)d_in[4];
  const float* W   = (const float*)d_in[5];
  const float* b   = (const float*)d_in[6];
  float* out = (float*)d_out;

  // workspace layout (doubles first for 8B alignment): ~8.2 KB total
  double*   S1    = (double*)d_ws;
  double*   SL    = S1 + NROW;
  double*   S0    = SL + NROW;
  double*   Ls    = S0 + NROW;
  float*    qnorm = (float*)(Ls + 1);
  float*    wd    = qnorm + NROW;
  float*    bdiff = wd + DIM;
  unsigned* minU  = (unsigned*)(bdiff + 1);
  unsigned* maxU  = minU + NROW;
  unsigned* cnt   = maxU + NROW;

  hipLaunchKernelGGL(init_kernel, dim3(1), dim3(256), 0, stream,
                     vl, W, b, S1, SL, S0, Ls, qnorm, wd, bdiff, minU, maxU, cnt);
  hipLaunchKernelGGL(sim_kernel, dim3(NTOT / 16), dim3(256), 0, stream,
                     vl, sc, itm, exq, exl, qnorm, wd, bdiff,
                     S1, SL, S0, Ls, minU, maxU, cnt);
  hipLaunchKernelGGL(fin_kernel, dim3(1), dim3(3*BSZ), 0, stream,
                     S1, SL, S0, Ls, minU, maxU, cnt, out);
}